// Energy_68015102099476
// MI455X (gfx1250) — compile-verified
//
#include <hip/hip_runtime.h>
#include <stdint.h>

// Problem constants (match reference).
#define BB 32
#define TT 500
#define DD 2048
#define SS 2048
#define EE 65536
#define NTHREADS 1024
#define LOG_LEAKY (-2.3025850929940457f)   // log(0.1)
#define NEG_INF_C (-1.0e30f)

// Pointer/vector types matching the async-to-LDS builtin signature:
//   void __builtin_amdgcn_global_load_async_to_lds_b64(v2i AS1*, v2i AS3*, imm int, imm int)
typedef int v2i_t __attribute__((ext_vector_type(2)));
typedef v2i_t __attribute__((address_space(1)))* gptr_v2i;
typedef v2i_t __attribute__((address_space(3)))* lptr_v2i;

// ---- order-preserving float<->uint map so float-max == ds_max_u32 ----
__device__ __forceinline__ uint32_t f2ou(float f) {
  int32_t b = __float_as_int(f);
  return (b >= 0) ? ((uint32_t)b ^ 0x80000000u) : ~(uint32_t)b;
}
__device__ __forceinline__ float ou2f(uint32_t u) {
  int32_t b = (u & 0x80000000u) ? (int32_t)(u ^ 0x80000000u) : (int32_t)(~u);
  return __int_as_float(b);
}

__device__ __forceinline__ float logaddexpf_(float a, float b) {
  float mx = fmaxf(a, b), mn = fminf(a, b);
  return mx + log1pf(__expf(mn - mx));
}

__device__ __forceinline__ float waveReduceMax(float v) {
  #pragma unroll
  for (int off = 16; off > 0; off >>= 1) v = fmaxf(v, __shfl_xor(v, off, 32));
  return v;
}
__device__ __forceinline__ float waveReduceSum(float v) {
  #pragma unroll
  for (int off = 16; off > 0; off >>= 1) v += __shfl_xor(v, off, 32);
  return v;
}

// ---- CDNA5 async global->LDS copy (8 bytes / lane) + split wait counter ----
__device__ __forceinline__ void async_copy_b64(const float* gsrc, float* ldst) {
#if __has_builtin(__builtin_amdgcn_global_load_async_to_lds_b64)
  __builtin_amdgcn_global_load_async_to_lds_b64(
      (gptr_v2i)(uintptr_t)gsrc,
      (lptr_v2i)(uint32_t)(uintptr_t)ldst,   // low 32 bits of flat LDS addr = LDS offset
      0, 0);
#else
  uint32_t lofs = (uint32_t)(uintptr_t)ldst;
  asm volatile("global_load_async_to_lds_b64 %0, %1, off"
               :: "v"(lofs), "v"(gsrc) : "memory");
#endif
}
__device__ __forceinline__ void wait_async0() {
#if __has_builtin(__builtin_amdgcn_s_wait_asynccnt)
  __builtin_amdgcn_s_wait_asynccnt(0);
#else
  asm volatile("s_wait_asynccnt 0" ::: "memory");
#endif
}

// ---- pack arcs into int4 so the hot loop does one b128 load per arc ----
__global__ __launch_bounds__(256) void pack_arcs_kernel(
    const int* __restrict__ fs, const int* __restrict__ ts,
    const int* __restrict__ pi, const float* __restrict__ lw,
    int4* __restrict__ packed) {
  int e = blockIdx.x * blockDim.x + threadIdx.x;
  if (e < EE) {
    int4 a;
    a.x = fs[e]; a.y = ts[e]; a.z = pi[e]; a.w = __float_as_int(lw[e]);
    packed[e] = a;
  }
}

template <bool PACKED>
__global__ __launch_bounds__(NTHREADS) void chain_forward_kernel(
    const float* __restrict__ x,           // [B,T,D]
    const int4*  __restrict__ arcs,        // [E] packed (if PACKED)
    const int*   __restrict__ fs, const int* __restrict__ ts,
    const int*   __restrict__ pi, const float* __restrict__ lw,
    const float* __restrict__ log_init,    // [S]
    const float* __restrict__ log_final,   // [S]
    float* __restrict__ objf_out)          // [B]
{
  __shared__ float    alpha[SS];
  __shared__ uint32_t mseg[SS];     // ordered-uint segment max
  __shared__ float    sseg[SS];     // segment sum of exp(score - m)
  __shared__ float    xt[2][DD];    // double-buffered frame of x
  __shared__ float    linit[SS];
  __shared__ uint32_t red_max_u;
  __shared__ float    red_sum;

  const int tid = threadIdx.x;
  const int b   = blockIdx.x;
  const float* xb = x + (size_t)b * TT * DD;

  for (int i = tid; i < SS; i += NTHREADS) {
    float li = log_init[i];
    linit[i] = li;
    alpha[i] = li;                          // alpha0 = log_init
  }
  if (tid == 0) { red_max_u = f2ou(-3.0e38f); red_sum = 0.f; }

  // prologue: stage frame 0 via async copy
  async_copy_b64(xb + 2 * tid, &xt[0][2 * tid]);
  wait_async0();
  __syncthreads();

  for (int t = 0; t < TT; ++t) {
    const int cur = t & 1;
    if (t + 1 < TT)  // prefetch next frame into the other buffer
      async_copy_b64(xb + (size_t)(t + 1) * DD + 2 * tid, &xt[cur ^ 1][2 * tid]);
    const float* xcur = xt[cur];

    // ---- tot = logsumexp_s(alpha) ----
    float a0 = alpha[tid], a1 = alpha[tid + NTHREADS];
    float lm = waveReduceMax(fmaxf(a0, a1));
    if ((tid & 31) == 0) atomicMax(&red_max_u, f2ou(lm));
    __syncthreads();
    float amax = ou2f(red_max_u);
    float p = __expf(a0 - amax) + __expf(a1 - amax);
    p = waveReduceSum(p);
    if ((tid & 31) == 0) atomicAdd(&red_sum, p);
    __syncthreads();
    float tot = amax + __logf(red_sum);
    __syncthreads();   // everyone has read red cells before reset

    // ---- leaky update + init segment accumulators ----
    alpha[tid]            = logaddexpf_(a0, LOG_LEAKY + linit[tid] + tot);
    alpha[tid + NTHREADS] = logaddexpf_(a1, LOG_LEAKY + linit[tid + NTHREADS] + tot);
    mseg[tid] = f2ou(NEG_INF_C); mseg[tid + NTHREADS] = f2ou(NEG_INF_C);
    sseg[tid] = 0.f;             sseg[tid + NTHREADS] = 0.f;
    if (tid == 0) { red_max_u = f2ou(-3.0e38f); red_sum = 0.f; }
    __syncthreads();

    // ---- pass 1: per-destination-state max over arcs ----
    for (int e = tid; e < EE; e += NTHREADS) {
      int frm, to, pdf; float w;
      if (PACKED) { int4 a = arcs[e]; frm = a.x; to = a.y; pdf = a.z; w = __int_as_float(a.w); }
      else        { frm = fs[e]; to = ts[e]; pdf = pi[e]; w = lw[e]; }
      float score = alpha[frm] + w + xcur[pdf];
      atomicMax(&mseg[to], f2ou(score));
    }
    __syncthreads();

    // ---- pass 2: sum exp(score - m[to]) ----
    for (int e = tid; e < EE; e += NTHREADS) {
      int frm, to, pdf; float w;
      if (PACKED) { int4 a = arcs[e]; frm = a.x; to = a.y; pdf = a.z; w = __int_as_float(a.w); }
      else        { frm = fs[e]; to = ts[e]; pdf = pi[e]; w = lw[e]; }
      float score = alpha[frm] + w + xcur[pdf];
      atomicAdd(&sseg[to], __expf(score - ou2f(mseg[to])));
    }
    __syncthreads();

    // ---- new alpha = m + log(max(sum, 1e-30)) ----
    alpha[tid]            = ou2f(mseg[tid]) + __logf(fmaxf(sseg[tid], 1e-30f));
    alpha[tid + NTHREADS] = ou2f(mseg[tid + NTHREADS]) + __logf(fmaxf(sseg[tid + NTHREADS], 1e-30f));
    wait_async0();     // this wave's prefetch has landed in LDS
    __syncthreads();   // all waves' prefetches visible
  }

  // ---- objf_b = logsumexp(alpha_T + log_final) ----
  float f0 = alpha[tid] + log_final[tid];
  float f1 = alpha[tid + NTHREADS] + log_final[tid + NTHREADS];
  float lm = waveReduceMax(fmaxf(f0, f1));
  if ((tid & 31) == 0) atomicMax(&red_max_u, f2ou(lm));
  __syncthreads();
  float amax = ou2f(red_max_u);
  float p = __expf(f0 - amax) + __expf(f1 - amax);
  p = waveReduceSum(p);
  if ((tid & 31) == 0) atomicAdd(&red_sum, p);
  __syncthreads();
  if (tid == 0) objf_out[b] = amax + __logf(red_sum);
}

__global__ __launch_bounds__(32) void finalize_kernel(const float* __restrict__ objf,
                                                      float* __restrict__ out) {
  float v = (threadIdx.x < BB) ? objf[threadIdx.x] : 0.f;
  v = waveReduceSum(v);
  if (threadIdx.x == 0) out[0] = -v;    // Energy returns -sum(objf)
}

extern "C" void kernel_launch(void* const* d_in, const int* in_sizes, int n_in,
                              void* d_out, int out_size, void* d_ws, size_t ws_size,
                              hipStream_t stream) {
  (void)in_sizes; (void)n_in; (void)out_size;
  const float* x  = (const float*)d_in[0];
  const float* lw = (const float*)d_in[1];
  const float* li = (const float*)d_in[2];
  const float* lf = (const float*)d_in[3];
  const int*   fs = (const int*)d_in[4];
  const int*   ts = (const int*)d_in[5];
  const int*   pi = (const int*)d_in[6];

  float* objf  = (float*)d_ws;                       // BB floats
  int4* packed = (int4*)((char*)d_ws + 256);         // EE int4
  const size_t need = 256 + (size_t)EE * sizeof(int4);

  if (ws_size >= need) {
    pack_arcs_kernel<<<EE / 256, 256, 0, stream>>>(fs, ts, pi, lw, packed);
    chain_forward_kernel<true><<<BB, NTHREADS, 0, stream>>>(
        x, packed, fs, ts, pi, lw, li, lf, objf);
  } else {
    chain_forward_kernel<false><<<BB, NTHREADS, 0, stream>>>(
        x, nullptr, fs, ts, pi, lw, li, lf, objf);
  }
  finalize_kernel<<<1, 32, 0, stream>>>(objf, (float*)d_out);
}